// PNet_9448928051825
// MI455X (gfx1250) — compile-verified
//
#include <hip/hip_runtime.h>

// ---------------------------------------------------------------------------
// PNet cascade (19 tiny dense layers) on MI455X via V_WMMA_F32_16X16X4_F32.
//
// Per layer:  D(dout x 16batch) = W^T * X^T, K chunked by 4, bias folded into
// W as an extra K-row paired with a constant-1.0 activation feature.
// A operand  = weight chunk (16x4 f32, 2 VGPRs), resident in registers (43x2).
// B operand  = transposed activations (4x16 f32, 2 VGPRs), built from the
//              previous D via lane-half swaps (v_permlanex16) + selects.
// C/D        = f32 16x16 accumulator (8 VGPRs); relu = single v_max_i32.
//
// Operand layouts (ISA 7.12.2):
//   A(m,k): lane = m + 16*(k>>1), vgpr = k&1            (documented)
//   B(k,n): lane = n + 16*(k>>1), vgpr = k&1            (mirror of A)
//   D(m,n): vgpr = m&7, lane = n + 16*(m>>3)            (documented)
// ---------------------------------------------------------------------------

typedef float v2f __attribute__((ext_vector_type(2)));
typedef float v8f __attribute__((ext_vector_type(8)));

#if defined(__HIP_DEVICE_COMPILE__)
#if !__has_builtin(__builtin_amdgcn_wmma_f32_16x16x4_f32)
#error "device pass: __builtin_amdgcn_wmma_f32_16x16x4_f32 not available"
#endif
#endif

static __device__ __forceinline__ v8f wmma4(v8f c, v2f a, v2f b) {
  // (neg_a, A, neg_b, B, c_mod, C, reuse_a, reuse_b)
  return __builtin_amdgcn_wmma_f32_16x16x4_f32(false, a, false, b, (short)0, c,
                                               false, false);
}

// Swap lane halves (lane ^ 16). v_permlanex16_b32 with identity selects is a
// pure-VALU half swap (no DS counter / s_wait_dscnt on the critical chain).
static __device__ __forceinline__ float swaph(float v) {
#if defined(__HIP_DEVICE_COMPILE__) && __has_builtin(__builtin_amdgcn_permlanex16)
  const int i = __float_as_int(v);
  return __int_as_float(
      __builtin_amdgcn_permlanex16(i, i, 0x76543210, 0xfedcba98, false, false));
#else
  // Fallback: ds_swizzle group-of-32, xor=0x10, and=0x1f (SWAPX16).
  return __int_as_float(__builtin_amdgcn_ds_swizzle(__float_as_int(v), 0x401f));
#endif
}

// Select per lane half: lanes 0-15 take `lo`, lanes 16-31 take `hiv`.
static __device__ __forceinline__ float mixh(bool hi, float lo, float hiv) {
  return hi ? hiv : lo;
}

static __device__ __forceinline__ v2f mkb(float a, float b) {
  v2f r; r.x = a; r.y = b; return r;
}

// Single-instruction ReLU via v_max_i32 on the raw bits: int ordering matches
// float ordering here (negatives incl. -0.0 have int-bits < 0 -> clamp to +0).
static __device__ __forceinline__ float relu1(float x) {
  int i = __float_as_int(x);
  i = i > 0 ? i : 0;
  return __int_as_float(i);
}

template <int N>
static __device__ __forceinline__ void relu(v8f& d) {
#pragma unroll
  for (int i = 0; i < N; ++i) d[i] = relu1(d[i]);
}

// ---------------------------------------------------------------------------
// Layer table:     fu1 fu2 ku1 ku2 hu1 hu2 md1 md2 md3 hd1 hd2 kd1 kd2 ha1 ha2 ka1 ka2 fa1 fa2
// din:              2   2   4   4   6   6  11  16  16   6   4   4   2   8   4   6   4   4   4
// dout:             2   2   4   4   6   6  16  16   6   4   4   2   2   4   1   4   1   4   1
// chunks=(din+4)>>2:1   1   2   2   2   2   3   5   5   2   2   2   1   3   2   2   2   2   2   = 43
// chunk base:       0   1   2   4   6   8  10  13  18  23  25  27  29  30  33  35  37  39  41
// ---------------------------------------------------------------------------

struct WPtrs {
  const float* w[19];
  const float* b[19];
};

// Pack all 43 weight chunks (bias folded as K-row din) into the WMMA A-operand
// lane layout, pairwise per lane for b64 loads: ws[cc*64 + lane*2 + v].
__global__ void pnet_prep(WPtrs p, float* __restrict__ wsA) {
  constexpr int LDIN[19]  = {2, 2, 4, 4, 6, 6, 11, 16, 16, 6, 4, 4, 2, 8, 4, 6, 4, 4, 4};
  constexpr int LDOUT[19] = {2, 2, 4, 4, 6, 6, 16, 16, 6, 4, 4, 2, 2, 4, 1, 4, 1, 4, 1};

  const int cc = blockIdx.x;      // global chunk id, 0..42
  const int lane = threadIdx.x;   // 0..31

  int l = 0, base = 0;
  for (int i = 0; i < 19; ++i) {
    int ch = (LDIN[i] + 4) >> 2;
    if (cc < base + ch) { l = i; break; }
    base += ch;
  }
  const int c = cc - base;
  const int din = LDIN[l], dout = LDOUT[l];
  const int m = lane & 15;

  for (int v = 0; v < 2; ++v) {
    const int k = 4 * c + v + 2 * (lane >> 4);   // A vgpr v, this lane -> K index
    float val = 0.0f;
    if (m < dout) {
      if (k < din)       val = p.w[l][k * dout + m];  // A[m][k] = W^T = w[k][m]
      else if (k == din) val = p.b[l][m];             // folded bias row
    }
    wsA[cc * 64 + lane * 2 + v] = val;
  }
}

__global__ __launch_bounds__(256) void pnet_wmma(
    const float* __restrict__ x, const float* __restrict__ wsA,
    float* __restrict__ out, int nrows) {
  const int lane = threadIdx.x & 31;
  const bool hi  = (threadIdx.x & 16) != 0;
  const int wave   = blockIdx.x * (blockDim.x >> 5) + (threadIdx.x >> 5);
  const int nwaves = gridDim.x * (blockDim.x >> 5);
  const int ntiles = (nrows + 15) >> 4;

  // Resident weights: 43 chunks, one b64 load each (L2-hot, once per wave).
  v2f W[43];
#pragma unroll
  for (int cc = 0; cc < 43; ++cc) {
    W[cc] = *reinterpret_cast<const v2f*>(wsA + cc * 64 + lane * 2);
  }

  for (int t = wave; t < ntiles; t += nwaves) {
    const int row = t * 16 + (lane & 15);
    const int rc  = row < nrows ? row : (nrows - 1);   // clamp tail tile loads
    const long rl = (long)rc * 11;
    // Raw columns duplicated into both lane halves (lanes l and l+16 same row)
    const float x0 = x[rl + 0], x1 = x[rl + 1], x2 = x[rl + 2], x3 = x[rl + 3];
    const float x4 = x[rl + 4], x5 = x[rl + 5], x6 = x[rl + 6], x7 = x[rl + 7];
    const float x8 = x[rl + 8], x9 = x[rl + 9], x10 = x[rl + 10];
    // h=x2 k=x3 f=x4 hd=x8 kd=x9 fd=x10, m_obs=[x0,x1,x5,x6,x7]

    // fu1: [f, fd, bias]                                    chunk 0
    v8f U = {};
    U = wmma4(U, W[0], mkb(mixh(hi, x4, 1.0f), x10));
    relu<2>(U);
    // fu2: [U0, U1, bias]                                   chunk 1
    v8f F = {};
    F = wmma4(F, W[1], mkb(mixh(hi, U[0], 1.0f), U[1]));
    relu<2>(F);                                              // f_up

    // ku1: [k, kd, F0, F1 | bias]                           chunks 2,3
    v8f K1 = {};
    K1 = wmma4(K1, W[2], mkb(mixh(hi, x3, swaph(F[0])), mixh(hi, x9, swaph(F[1]))));
    K1 = wmma4(K1, W[3], mkb(1.0f, 1.0f));
    relu<4>(K1);
    // ku2: [K1_0..3 | bias]                                 chunks 4,5
    v8f KU = {};
    KU = wmma4(KU, W[4], mkb(mixh(hi, K1[0], swaph(K1[2])), mixh(hi, K1[1], swaph(K1[3]))));
    KU = wmma4(KU, W[5], mkb(1.0f, 1.0f));
    relu<4>(KU);                                             // k_up

    // hu1: [h, hd, KU0, KU1 | KU2, KU3, bias]               chunks 6,7
    v8f H1 = {};
    H1 = wmma4(H1, W[6], mkb(mixh(hi, x2, swaph(KU[0])), mixh(hi, x8, swaph(KU[1]))));
    H1 = wmma4(H1, W[7], mkb(mixh(hi, KU[2], 1.0f), KU[3]));
    relu<6>(H1);
    // hu2: [H1_0..3 | H1_4, H1_5, bias]                     chunks 8,9
    v8f HU = {};
    HU = wmma4(HU, W[8], mkb(mixh(hi, H1[0], swaph(H1[2])), mixh(hi, H1[1], swaph(H1[3]))));
    HU = wmma4(HU, W[9], mkb(mixh(hi, H1[4], 1.0f), H1[5]));
    relu<6>(HU);                                             // h_up

    // md1: [x0,x1,x5,x6 | x7,HU0,HU1,HU2 | HU3,HU4,HU5,bias]  chunks 10..12
    v8f M1 = {};
    M1 = wmma4(M1, W[10], mkb(mixh(hi, x0, x5), mixh(hi, x1, x6)));
    M1 = wmma4(M1, W[11], mkb(mixh(hi, x7, swaph(HU[1])), mixh(hi, HU[0], swaph(HU[2]))));
    M1 = wmma4(M1, W[12], mkb(mixh(hi, HU[3], swaph(HU[5])), mixh(hi, HU[4], 1.0f)));
    relu<8>(M1);                                             // dout=16, both halves
    // md2: [M1_0..15 | bias]                                chunks 13..17
    v8f M2 = {};
    M2 = wmma4(M2, W[13], mkb(mixh(hi, M1[0], swaph(M1[2])), mixh(hi, M1[1], swaph(M1[3]))));
    M2 = wmma4(M2, W[14], mkb(mixh(hi, M1[4], swaph(M1[6])), mixh(hi, M1[5], swaph(M1[7]))));
    M2 = wmma4(M2, W[15], mkb(mixh(hi, swaph(M1[0]), M1[2]), mixh(hi, swaph(M1[1]), M1[3])));
    M2 = wmma4(M2, W[16], mkb(mixh(hi, swaph(M1[4]), M1[6]), mixh(hi, swaph(M1[5]), M1[7])));
    M2 = wmma4(M2, W[17], mkb(1.0f, 1.0f));
    relu<8>(M2);
    // md3: [M2_0..15 | bias]                                chunks 18..22
    v8f MD = {};
    MD = wmma4(MD, W[18], mkb(mixh(hi, M2[0], swaph(M2[2])), mixh(hi, M2[1], swaph(M2[3]))));
    MD = wmma4(MD, W[19], mkb(mixh(hi, M2[4], swaph(M2[6])), mixh(hi, M2[5], swaph(M2[7]))));
    MD = wmma4(MD, W[20], mkb(mixh(hi, swaph(M2[0]), M2[2]), mixh(hi, swaph(M2[1]), M2[3])));
    MD = wmma4(MD, W[21], mkb(mixh(hi, swaph(M2[4]), M2[6]), mixh(hi, swaph(M2[5]), M2[7])));
    MD = wmma4(MD, W[22], mkb(1.0f, 1.0f));
    relu<6>(MD);                                             // m_down

    // hd1: [MD0..3 | MD4, MD5, bias]                        chunks 23,24
    v8f D1 = {};
    D1 = wmma4(D1, W[23], mkb(mixh(hi, MD[0], swaph(MD[2])), mixh(hi, MD[1], swaph(MD[3]))));
    D1 = wmma4(D1, W[24], mkb(mixh(hi, MD[4], 1.0f), MD[5]));
    relu<4>(D1);
    // hd2: [D1_0..3 | bias]                                 chunks 25,26
    v8f HD = {};
    HD = wmma4(HD, W[25], mkb(mixh(hi, D1[0], swaph(D1[2])), mixh(hi, D1[1], swaph(D1[3]))));
    HD = wmma4(HD, W[26], mkb(1.0f, 1.0f));
    relu<4>(HD);                                             // h_down

    // kd1: [HD0..3 | bias]                                  chunks 27,28
    v8f KD1 = {};
    KD1 = wmma4(KD1, W[27], mkb(mixh(hi, HD[0], swaph(HD[2])), mixh(hi, HD[1], swaph(HD[3]))));
    KD1 = wmma4(KD1, W[28], mkb(1.0f, 1.0f));
    relu<2>(KD1);
    // kd2: [KD1_0, KD1_1, bias]                             chunk 29
    v8f KDD = {};
    KDD = wmma4(KDD, W[29], mkb(mixh(hi, KD1[0], 1.0f), KD1[1]));
    relu<2>(KDD);                                            // k_down

    // ha1: [h, hd, MD0, MD1 | MD2..MD5 | bias]              chunks 30..32
    v8f A1 = {};
    A1 = wmma4(A1, W[30], mkb(mixh(hi, x2, swaph(MD[0])), mixh(hi, x8, swaph(MD[1]))));
    A1 = wmma4(A1, W[31], mkb(mixh(hi, MD[2], swaph(MD[4])), mixh(hi, MD[3], swaph(MD[5]))));
    A1 = wmma4(A1, W[32], mkb(1.0f, 1.0f));
    relu<4>(A1);
    // ha2: [A1_0..3 | bias], linear                         chunks 33,34
    v8f HA = {};
    HA = wmma4(HA, W[33], mkb(mixh(hi, A1[0], swaph(A1[2])), mixh(hi, A1[1], swaph(A1[3]))));
    HA = wmma4(HA, W[34], mkb(1.0f, 1.0f));
    const float h_act = HA[0];

    // ka1: [k, kd, HD0, HD1 | HD2, HD3, bias]               chunks 35,36
    v8f A2 = {};
    A2 = wmma4(A2, W[35], mkb(mixh(hi, x3, swaph(HD[0])), mixh(hi, x9, swaph(HD[1]))));
    A2 = wmma4(A2, W[36], mkb(mixh(hi, HD[2], 1.0f), HD[3]));
    relu<4>(A2);
    // ka2: [A2_0..3 | bias], linear                         chunks 37,38
    v8f KA = {};
    KA = wmma4(KA, W[37], mkb(mixh(hi, A2[0], swaph(A2[2])), mixh(hi, A2[1], swaph(A2[3]))));
    KA = wmma4(KA, W[38], mkb(1.0f, 1.0f));
    const float k_act = KA[0];

    // fa1: [f, fd, KDD0, KDD1 | bias]                       chunks 39,40
    v8f A3 = {};
    A3 = wmma4(A3, W[39], mkb(mixh(hi, x4, swaph(KDD[0])), mixh(hi, x10, swaph(KDD[1]))));
    A3 = wmma4(A3, W[40], mkb(1.0f, 1.0f));
    relu<4>(A3);
    // fa2: [A3_0..3 | bias], linear                         chunks 41,42
    v8f FA = {};
    FA = wmma4(FA, W[41], mkb(mixh(hi, A3[0], swaph(A3[2])), mixh(hi, A3[1], swaph(A3[3]))));
    FA = wmma4(FA, W[42], mkb(1.0f, 1.0f));
    const float f_act = FA[0];

    // out[:, 0]=h_act, [:,1]=f_act, [:,2]=k_act; row-major, lanes 0-15 only.
    if (lane < 16 && row < nrows) {
      const long o = (long)row * 3;
      out[o + 0] = h_act;
      out[o + 1] = f_act;
      out[o + 2] = k_act;
    }
  }
}

extern "C" void kernel_launch(void* const* d_in, const int* in_sizes, int n_in,
                              void* d_out, int out_size, void* d_ws, size_t ws_size,
                              hipStream_t stream) {
  const float* x = (const float*)d_in[0];
  WPtrs p;
  for (int i = 0; i < 19; ++i) {
    p.w[i] = (const float*)d_in[1 + 2 * i];
    p.b[i] = (const float*)d_in[2 + 2 * i];
  }
  float* wsA = (float*)d_ws;   // 43 chunks * 64 floats = 11 KB
  float* out = (float*)d_out;
  const int nrows = in_sizes[0] / 11;

  pnet_prep<<<43, 32, 0, stream>>>(p, wsA);

  const int ntiles = (nrows + 15) >> 4;
  const int wavesPerBlock = 8;            // 256 threads = 8 wave32
  int blocks = (ntiles + wavesPerBlock - 1) / wavesPerBlock;
  if (blocks > 1024) blocks = 1024;       // grid-stride; ~15 tiles/wave at B=2M
  pnet_wmma<<<blocks, 256, 0, stream>>>(x, wsA, out, nrows);
}